// DistMultPredictor_26310969655543
// MI455X (gfx1250) — compile-verified
//
#include <hip/hip_runtime.h>
#include <math.h>

// DistMult edge scoring, CDNA5 (gfx1250), wave32.
//
// score[e] = sigmoid( sum_k h[src[e],k] * W[rel[e],k] * h[dst[e],k] )
//
// Strategy: per wave, take a tile of 16 edges. A[m,k] = h_u[m,k]*h_v[m,k]
// (16x128), B = W^T zero-padded to 128x16. D = A x B via 32 chained
// V_WMMA_F32_16X16X4_F32 -> D[m,n] = score of edge m under relation n.
// Extract D[m, rel[m]] through a per-wave LDS stage, apply sigmoid.
// Gather-bound kernel (h fits in 192MB L2); WMMA does the K=128 reduction
// for free on the otherwise-idle matrix pipe.

typedef __attribute__((ext_vector_type(2))) float v2f;
typedef __attribute__((ext_vector_type(8))) float v8f;

#define D_DIM 128
#define NRELS 6
#define WAVES_PER_BLOCK 8
#define EDGES_PER_WAVE 16
#define BLOCK_THREADS (WAVES_PER_BLOCK * 32)

__global__ __launch_bounds__(BLOCK_THREADS)
void distmult_wmma_kernel(const float* __restrict__ h,
                          const float* __restrict__ W,
                          const int*   __restrict__ src,
                          const int*   __restrict__ dst,
                          const int*   __restrict__ rel,
                          float*       __restrict__ out,
                          int E)
{
    // W^T staged in LDS, zero-padded to 16 "relations" so columns 6..15 of the
    // WMMA B operand are 0 (no divergence, no garbage in unused columns).
    __shared__ float ldsW[16 * D_DIM];                  // 8 KB
    __shared__ float ldsD[WAVES_PER_BLOCK * 16 * 16];   // 8 KB, D-tile staging

    const int tid = threadIdx.x;

    // Cooperative preload: 2048 floats / 256 threads = 8 each.
    #pragma unroll
    for (int i = 0; i < (16 * D_DIM) / BLOCK_THREADS; ++i) {
        int idx = tid + i * BLOCK_THREADS;
        ldsW[idx] = (idx < NRELS * D_DIM) ? W[idx] : 0.0f;
    }
    __syncthreads();

    const int wave = tid >> 5;        // wave32
    const int lane = tid & 31;
    const int m    = lane & 15;       // A row / D column owned by this lane
    const int kh   = lane >> 4;       // 0: K pairs {4k,4k+1}; 1: {4k+2,4k+3}

    const long tile = (long)blockIdx.x * WAVES_PER_BLOCK + wave;
    const long e    = tile * EDGES_PER_WAVE + m;
    const long ec   = (e < (long)E) ? e : (long)(E - 1);  // clamp: keep EXEC all-1s

    const int s = src[ec];
    const int d = dst[ec];
    const int r = rel[ec];

    // Per-lane streaming bases; step k reads bytes [16k, 16k+8).
    const float* pu = h + (size_t)s * D_DIM + 2 * kh;
    const float* pv = h + (size_t)d * D_DIM + 2 * kh;
    const float* pb = ldsW + (size_t)m * D_DIM + 2 * kh;

    v8f acc = {};
    #pragma unroll
    for (int k = 0; k < D_DIM / 4; ++k) {
        v2f u2 = *(const v2f*)(pu + 4 * k);   // global_load_b64 (L2 gather)
        v2f v2 = *(const v2f*)(pv + 4 * k);   // global_load_b64 (L2 gather)
        v2f b2 = *(const v2f*)(pb + 4 * k);   // ds_load_b64 (W^T)
        v2f a;
        a.x = u2.x * v2.x;                    // A[m,4k+2h]   = h_u * h_v
        a.y = u2.y * v2.y;                    // A[m,4k+2h+1]
        // D += A(16x4) x B(4x16), full f32
        acc = __builtin_amdgcn_wmma_f32_16x16x4_f32(
            false, a, false, b2, (short)0, acc, false, false);
    }

    // D layout: acc[i] holds D[i + 8*kh, m]. Stage the 16x16 tile to LDS so
    // each edge-row lane can pick its rel-dependent column.
    float* wd = ldsD + wave * 256;
    #pragma unroll
    for (int i = 0; i < 8; ++i)
        wd[(i + 8 * kh) * 16 + m] = acc[i];
    __syncthreads();   // also orders the per-wave LDS store->load lane crossing

    if (kh == 0 && e < (long)E) {
        float sc = wd[m * 16 + r];            // D[m, rel[m]]
        out[e] = 1.0f / (1.0f + __expf(-sc)); // sigmoid
    }
}

extern "C" void kernel_launch(void* const* d_in, const int* in_sizes, int n_in,
                              void* d_out, int out_size, void* d_ws, size_t ws_size,
                              hipStream_t stream) {
    const float* h   = (const float*)d_in[0];
    const float* W   = (const float*)d_in[1];
    const int*   src = (const int*)d_in[2];
    const int*   dst = (const int*)d_in[3];
    const int*   rel = (const int*)d_in[4];
    float*       out = (float*)d_out;

    const int E      = in_sizes[2];                           // 1,500,000
    const int tiles  = (E + EDGES_PER_WAVE - 1) / EDGES_PER_WAVE;
    const int blocks = (tiles + WAVES_PER_BLOCK - 1) / WAVES_PER_BLOCK;

    distmult_wmma_kernel<<<blocks, BLOCK_THREADS, 0, stream>>>(
        h, W, src, dst, rel, out, E);
}